// SpikingTactile_48842368090340
// MI455X (gfx1250) — compile-verified
//
#include <hip/hip_runtime.h>
#include <stdint.h>

#define STEPS 10
#define NNEU  6
#define TPB   256

#if __has_builtin(__builtin_amdgcn_tensor_load_to_lds) && __has_builtin(__builtin_amdgcn_s_wait_tensorcnt)
#define USE_TDM 1
#endif

#if __has_builtin(__builtin_amdgcn_rcpf)
__device__ __forceinline__ float fast_rcp(float x) { return __builtin_amdgcn_rcpf(x); }
#else
__device__ __forceinline__ float fast_rcp(float x) { return 1.0f / x; }
#endif

typedef uint32_t u32x4 __attribute__((ext_vector_type(4)));
typedef int32_t  i32x4 __attribute__((ext_vector_type(4)));
typedef int32_t  i32x8 __attribute__((ext_vector_type(8)));

// Touch logic + predictive-coding column (4 units/channel are identical -> compute 1).
// Fills o9[0..8] = {head,trunk,tail,intensity,contact,startle,pe,prec,fe} and I[0..5].
__device__ __forceinline__ void touch_pc(bool col, float wall, float h2w, float cons,
                                         float tb, float pred, float* o9, float* I)
{
  float head = (col && h2w > 0.3f) ? 0.8f
             : ((wall > 0.8f && h2w > 0.5f) ? 0.4f * wall : 0.0f);
  head += fmaxf(0.0f, (25.0f - pred) * (1.0f / 25.0f)) * 0.7f;
  head  = fminf(head, 1.0f);

  float trunk = fmaxf(0.0f, (20.0f - cons) * (1.0f / 20.0f)) * 0.6f * ((cons < 20.0f) ? 1.0f : 0.0f);
  trunk += (wall > 0.6f && fabsf(h2w) < 0.3f) ? 0.3f * wall : 0.0f;
  trunk += fmaxf(0.0f, (30.0f - pred) * (1.0f / 30.0f)) * 0.5f;
  trunk  = fminf(trunk, 1.0f);

  const float self_gen = tb * 0.3f;
  float external = (col && h2w < -0.3f) ? 0.5f : 0.0f;
  external += (h2w < 0.0f) ? fmaxf(0.0f, (35.0f - pred) * (1.0f / 35.0f)) * 0.4f : 0.0f;
  const float predicted_tail = self_gen;
  const float tail = fminf(1.0f, external + fmaxf(0.0f, self_gen - predicted_tail * 0.8f));

  const float intensity = fmaxf(fmaxf(head, trunk), tail);
  const bool hq = (head  >= trunk) && (head  >= tail) && (head  > 0.05f);
  const bool tq = (trunk >= head)  && (trunk >= tail) && (trunk > 0.05f);
  const float contact = hq ? 1.0f : (tq ? 2.0f : ((tail > 0.05f) ? 3.0f : 0.0f));
  const float startle = (head > 0.5f) ? 1.0f : 0.0f;

  float sch[3] = {head, trunk, tail};
  float pch[3] = {0.0f, 0.0f, predicted_tail};
  float mu[3]  = {0.0f, 0.0f, 0.0f};
  float pr[3]  = {1.0f, 1.0f, 1.0f};
#pragma unroll
  for (int k = 0; k < 8; ++k) {
#pragma unroll
    for (int c = 0; c < 3; ++c) {
      float err = sch[c] - mu[c] - pch[c];
      mu[c] += 0.1f * pr[c] * err - 0.01f * mu[c];
      pr[c] += 0.05f * (fast_rcp(err * err + 0.5f) - pr[c]);   // continuous outputs only
    }
  }
  float peA = 0.0f, prA = 0.0f, feA = 0.0f;
#pragma unroll
  for (int c = 0; c < 3; ++c) {
    float pe = sch[c] - mu[c] - pch[c];
    peA += fabsf(pe);
    prA += pr[c];
    feA += pr[c] * pe * pe - __logf(pr[c]);
  }

  o9[0] = head;      o9[1] = trunk;            o9[2] = tail;
  o9[3] = intensity; o9[4] = contact;          o9[5] = startle;
  o9[6] = peA * (1.0f / 3.0f);
  o9[7] = prA * (1.0f / 3.0f);
  o9[8] = 0.5f * feA * (1.0f / 3.0f);

  I[0] = head  * 15.0f - 1.0f;  I[1] = head  * 8.0f - 1.0f;
  I[2] = trunk * 12.0f - 1.0f;  I[3] = trunk * 8.0f - 1.0f;
  I[4] = tail  * 10.0f - 1.0f;  I[5] = tail  * 8.0f - 1.0f;
}

// 10-step Izhikevich RS recurrence; v/u pre-loaded in registers.
// Kept in the reference's evaluation order (spike threshold is discrete).
__device__ __forceinline__ float izhikevich_rate(const float* __restrict__ ns, int stride,
                                                 float* v, float* u, const float* I)
{
  float r[NNEU];
#pragma unroll
  for (int n = 0; n < NNEU; ++n) r[n] = 0.0f;
#pragma unroll
  for (int t = 0; t < STEPS; ++t) {
#pragma unroll
    for (int n = 0; n < NNEU; ++n) {
      float inp = I[n] + 0.3f * ns[(size_t)t * (size_t)stride + n];
      float vv  = v[n];
      float v2  = vv + (0.04f * vv * vv + 5.0f * vv + 140.0f - u[n] + inp);
      float u2  = u[n] + 0.02f * (0.2f * vv - u[n]);
      bool spk  = v2 >= 30.0f;
      v[n] = spk ? -65.0f : v2;
      u[n] = spk ? u2 + 8.0f : u2;
      r[n] = 0.95f * r[n] + (spk ? 0.05f : 0.0f);
    }
  }
  float s = 0.0f;
#pragma unroll
  for (int n = 0; n < NNEU; ++n) s += r[n];
  return s * (1.0f / (float)NNEU);
}

// Hot kernel: every block is full (256 agents). Branch-free TDM pipeline.
__global__ void __launch_bounds__(TPB)
spiking_tactile_main(const unsigned char* __restrict__ collision,
                     const float* __restrict__ wall_proximity,
                     const float* __restrict__ heading_to_wall,
                     const float* __restrict__ conspecific_distance,
                     const float* __restrict__ tail_beat_in,
                     const float* __restrict__ predator_distance,
                     const float* __restrict__ noise,
                     const float* __restrict__ v0,
                     const float* __restrict__ u0,
                     float* __restrict__ out,
                     int Btotal)
{
  const int tid   = (int)threadIdx.x;
  const int base  = (int)blockIdx.x * TPB;
  const int agent = base + tid;

#ifdef USE_TDM
  __shared__ float smem[TPB * STEPS * NNEU];   // 60 KB: [wave][step][lane][neuron]
  const int waveBase = __builtin_amdgcn_readfirstlane(tid);   // 32*waveId (uniform)
  {
    // One TDM descriptor per wave: 2D tile, rows = STEPS, row = 32 agents * 6 f32.
    const uint64_t gaddr  = (uint64_t)(uintptr_t)(noise + (size_t)(base + waveBase) * NNEU);
    const uint32_t ldsOff = (uint32_t)(uintptr_t)&smem[waveBase * (STEPS * NNEU)];
    u32x4 g0;
    g0[0] = 1u;                                     // count=1 (valid), user mode
    g0[1] = ldsOff;                                 // lds_addr (bytes)
    g0[2] = (uint32_t)gaddr;                        // global_addr[31:0]
    g0[3] = (uint32_t)(gaddr >> 32) | 0x80000000u;  // global_addr[56:32] | type=2
    const uint32_t td0   = 0x7FFFFFFFu;             // tensor_dim0: huge -> no OOB zero-fill
    const uint32_t td1   = 0x7FFFu;                 // tensor_dim1: huge -> no OOB zero-fill
    const uint32_t tile0 = 32u * NNEU;              // 192 elements per row
    const uint32_t tile1 = STEPS;                   // 10 rows
    const uint64_t s0    = (uint64_t)Btotal * NNEU; // stride between step planes (elems)
    i32x8 g1;
    g1[0] = (int)(2u << 16);                                  // mask=0 | data_size=4B
    g1[1] = (int)((td0 & 0xFFFFu) << 16);                     // atomic_addr=0 | td0.lo16
    g1[2] = (int)((td0 >> 16) | ((td1 & 0xFFFFu) << 16));     // td0.hi16 | td1.lo16
    g1[3] = (int)((td1 >> 16) | (tile0 << 16));               // td1.hi16 | tile_dim0
    g1[4] = (int)tile1;                                       // tile_dim1 | tile_dim2=0
    g1[5] = (int)(uint32_t)(s0 & 0xFFFFFFFFu);                // dim0_stride[31:0]
    g1[6] = (int)(uint32_t)((s0 >> 32) & 0xFFFFu);            // dim0_stride[47:32]
    g1[7] = 0;
    i32x4 gz = {0, 0, 0, 0};
#if __clang_major__ >= 23
    i32x8 gz8 = {0, 0, 0, 0, 0, 0, 0, 0};
    __builtin_amdgcn_tensor_load_to_lds(g0, g1, gz, gz, gz8, 0);
#else
    __builtin_amdgcn_tensor_load_to_lds(g0, g1, gz, gz, 0);
#endif
  }
#endif

  // ---- everything below overlaps with the TDM DMA ----
  const bool  col  = collision[agent] != 0;
  const float wall = wall_proximity[agent];
  const float h2w  = heading_to_wall[agent];
  const float cons = conspecific_distance[agent];
  const float tb   = tail_beat_in[agent];
  const float pred = predator_distance[agent];

  float o9[9], I[NNEU];
  touch_pc(col, wall, h2w, cons, tb, pred, o9, I);

  float v[NNEU], u[NNEU];
  const float* vp = v0 + (size_t)agent * NNEU;
  const float* up = u0 + (size_t)agent * NNEU;
#pragma unroll
  for (int n = 0; n < NNEU; ++n) { v[n] = vp[n]; u[n] = up[n]; }

  float rate_mean;
#ifdef USE_TDM
  __builtin_amdgcn_s_wait_tensorcnt(0);          // this wave's DMA complete
  const float* ns = &smem[waveBase * (STEPS * NNEU) + (tid - waveBase) * NNEU];
  asm volatile("" : : "v"(ns) : "memory");       // TDM wrote LDS behind the compiler's back
  rate_mean = izhikevich_rate(ns, 32 * NNEU, v, u, I);
#else
  rate_mean = izhikevich_rate(noise + (size_t)agent * NNEU, Btotal * NNEU, v, u, I);
#endif

  float* o = out + (size_t)agent * 10;
#pragma unroll
  for (int k = 0; k < 9; ++k) o[k] = o9[k];
  o[9] = rate_mean;
}

// Tail kernel: handles agents in [start, Btotal) when B % 256 != 0. Global noise path.
__global__ void __launch_bounds__(TPB)
spiking_tactile_tail(const unsigned char* __restrict__ collision,
                     const float* __restrict__ wall_proximity,
                     const float* __restrict__ heading_to_wall,
                     const float* __restrict__ conspecific_distance,
                     const float* __restrict__ tail_beat_in,
                     const float* __restrict__ predator_distance,
                     const float* __restrict__ noise,
                     const float* __restrict__ v0,
                     const float* __restrict__ u0,
                     float* __restrict__ out,
                     int Btotal, int start)
{
  const int agent = start + (int)blockIdx.x * TPB + (int)threadIdx.x;
  if (agent >= Btotal) return;

  const bool  col  = collision[agent] != 0;
  const float wall = wall_proximity[agent];
  const float h2w  = heading_to_wall[agent];
  const float cons = conspecific_distance[agent];
  const float tb   = tail_beat_in[agent];
  const float pred = predator_distance[agent];

  float o9[9], I[NNEU];
  touch_pc(col, wall, h2w, cons, tb, pred, o9, I);

  float v[NNEU], u[NNEU];
  const float* vp = v0 + (size_t)agent * NNEU;
  const float* up = u0 + (size_t)agent * NNEU;
#pragma unroll
  for (int n = 0; n < NNEU; ++n) { v[n] = vp[n]; u[n] = up[n]; }

  const float rate_mean =
      izhikevich_rate(noise + (size_t)agent * NNEU, Btotal * NNEU, v, u, I);

  float* o = out + (size_t)agent * 10;
#pragma unroll
  for (int k = 0; k < 9; ++k) o[k] = o9[k];
  o[9] = rate_mean;
}

extern "C" void kernel_launch(void* const* d_in, const int* in_sizes, int n_in,
                              void* d_out, int out_size, void* d_ws, size_t ws_size,
                              hipStream_t stream) {
  (void)n_in; (void)out_size; (void)d_ws; (void)ws_size;
  const int B = in_sizes[0];
  const unsigned char* collision = (const unsigned char*)d_in[0];  // jax bool -> 1 byte
  const float* wall  = (const float*)d_in[1];
  const float* h2w   = (const float*)d_in[2];
  const float* consp = (const float*)d_in[3];
  // d_in[4] = swim_speed: unused by the reference math
  const float* tb    = (const float*)d_in[5];
  const float* pred  = (const float*)d_in[6];
  const float* noise = (const float*)d_in[7];
  const float* v0    = (const float*)d_in[8];
  const float* u0    = (const float*)d_in[9];
  float* out = (float*)d_out;

  const int fullBlocks = B / TPB;
  const int tailStart  = fullBlocks * TPB;
  if (fullBlocks > 0) {
    spiking_tactile_main<<<fullBlocks, TPB, 0, stream>>>(collision, wall, h2w, consp, tb,
                                                         pred, noise, v0, u0, out, B);
  }
  if (tailStart < B) {
    const int tailBlocks = (B - tailStart + TPB - 1) / TPB;
    spiking_tactile_tail<<<tailBlocks, TPB, 0, stream>>>(collision, wall, h2w, consp, tb,
                                                         pred, noise, v0, u0, out, B, tailStart);
  }
}